// HGDCODE_32933809225802
// MI455X (gfx1250) — compile-verified
//
#include <hip/hip_runtime.h>
#include <math.h>

// ---------------------------------------------------------------------------
// Problem constants (from reference): B=16, T=48, N=400, DIN=2, R=H=64, K=3
// ---------------------------------------------------------------------------
#define BB 16
#define TT 48
#define NN 400
#define RR 64

typedef __attribute__((ext_vector_type(2))) float v2f;
typedef __attribute__((ext_vector_type(8))) float v8f;

__device__ __forceinline__ v8f wmma4(v2f a, v2f b, v8f c) {
  // V_WMMA_F32_16X16X4_F32 : exact fp32 matrix core path on gfx1250
  return __builtin_amdgcn_wmma_f32_16x16x4_f32(
      /*neg_a=*/false, a, /*neg_b=*/false, b,
      /*c_mod=*/(short)0, c, /*reuse_a=*/false, /*reuse_b=*/false);
}

__device__ __forceinline__ v8f vzero8() {
  v8f z = {0.f,0.f,0.f,0.f,0.f,0.f,0.f,0.f};
  return z;
}
__device__ __forceinline__ float sigmoidf_(float x) {
  return 1.0f / (1.0f + expf(-x));
}

// ---------------------------------------------------------------------------
// Diffusion GEMM:  Y[b] = A[b] (400x400) * X[b] (400xD), D in {64,96}
// one wave per 16xD strip; grid = (25, B, nState)
// ---------------------------------------------------------------------------
template<int D>
__global__ __launch_bounds__(32)
void k_diffuse(const float* __restrict__ A,
               const float* __restrict__ src0, long sb0,
               const float* __restrict__ src1, long sb1,
               float* __restrict__ dst0, long db0,
               float* __restrict__ dst1, long db1)
{
  const int lane = threadIdx.x;
  const int m0   = blockIdx.x * 16;
  const int b    = blockIdx.y;
  const int s    = blockIdx.z;
  const bool hi  = lane >= 16;
  const int  ml  = lane & 15;
  const int  koff = hi ? 2 : 0;

  const float* X = (s == 0) ? (src0 + (long)b * sb0) : (src1 + (long)b * sb1);
  float*       Y = (s == 0) ? (dst0 + (long)b * db0) : (dst1 + (long)b * db1);
  const float* Ab = A + (long)b * (NN * NN);

  v8f acc[D / 16];
  #pragma unroll
  for (int i = 0; i < D / 16; ++i) acc[i] = vzero8();

  const float* arow = Ab + (long)(m0 + ml) * NN;
  for (int k0 = 0; k0 < NN; k0 += 4) {
    v2f af = { arow[k0 + koff], arow[k0 + koff + 1] };
    const float* xr = X + (long)(k0 + koff) * D + ml;
    #pragma unroll
    for (int nt = 0; nt < D / 16; ++nt) {
      v2f bf = { xr[nt * 16], xr[nt * 16 + D] };
      acc[nt] = wmma4(af, bf, acc[nt]);
    }
  }
  #pragma unroll
  for (int nt = 0; nt < D / 16; ++nt)
    #pragma unroll
    for (int r = 0; r < 8; ++r)
      Y[(long)(m0 + r + (hi ? 8 : 0)) * D + nt * 16 + ml] = acc[nt][r];
}

// ---------------------------------------------------------------------------
// AX1[b][n][t*2+c] = sum_j A[b][n][j] * (values*masks)[b][t][j][c]   (D=96)
// ---------------------------------------------------------------------------
__global__ __launch_bounds__(32)
void k_ax_input(const float* __restrict__ A, const float* __restrict__ vals,
                const float* __restrict__ msk, float* __restrict__ AX1)
{
  const int lane = threadIdx.x;
  const int m0 = blockIdx.x * 16;
  const int b  = blockIdx.y;
  const bool hi = lane >= 16;
  const int ml = lane & 15;
  const int koff = hi ? 2 : 0;
  const float* Ab = A + (long)b * (NN * NN);

  v8f acc[6];
  #pragma unroll
  for (int i = 0; i < 6; ++i) acc[i] = vzero8();

  const float* arow = Ab + (long)(m0 + ml) * NN;
  for (int k0 = 0; k0 < NN; k0 += 4) {
    v2f af = { arow[k0 + koff], arow[k0 + koff + 1] };
    const int row = k0 + koff;
    #pragma unroll
    for (int nt = 0; nt < 6; ++nt) {
      const int col = nt * 16 + ml;
      const int t = col >> 1, c = col & 1;
      const long i0 = (((long)b * TT + t) * NN + row) * 2 + c;
      v2f bf = { vals[i0] * msk[i0], vals[i0 + 2] * msk[i0 + 2] };
      acc[nt] = wmma4(af, bf, acc[nt]);
    }
  }
  float* Y = AX1 + (long)b * (NN * 96);
  #pragma unroll
  for (int nt = 0; nt < 6; ++nt)
    #pragma unroll
    for (int r = 0; r < 8; ++r)
      Y[(long)(m0 + r + (hi ? 8 : 0)) * 96 + nt * 16 + ml] = acc[nt][r];
}

// ---------------------------------------------------------------------------
// ODE layer eval: out = [res +] tanh( S0*W[0] + S1*W[1] + S2*W[2] + b )
// K = 3*64 = 192, out 64 cols.  Two states per launch.
// ---------------------------------------------------------------------------
struct L3Args {
  const float* src[2][3];
  const float* W[2];
  const float* bias[2];
  const float* res[2];
  long         resB;
  float*       dst[2];
  long         dstB[2];
  long         srcB;
};

__global__ __launch_bounds__(32)
void k_layer3(L3Args a)
{
  const int lane = threadIdx.x;
  const int m0 = blockIdx.x * 16;
  const int b  = blockIdx.y;
  const int s  = blockIdx.z;
  const bool hi = lane >= 16;
  const int ml = lane & 15;
  const int koff = hi ? 2 : 0;
  const float* W = a.W[s];

  v8f acc[4];
  #pragma unroll
  for (int i = 0; i < 4; ++i) acc[i] = vzero8();

  for (int k0 = 0; k0 < 192; k0 += 4) {
    const int chunk = k0 >> 6, d0 = k0 & 63;
    const float* S = a.src[s][chunk] + (long)b * a.srcB;
    const float* sp = S + (long)(m0 + ml) * 64 + d0 + koff;
    v2f af = { sp[0], sp[1] };
    const float* wp = W + (long)chunk * 4096 + (long)(d0 + koff) * 64 + ml;
    #pragma unroll
    for (int nt = 0; nt < 4; ++nt) {
      v2f bf = { wp[nt * 16], wp[nt * 16 + 64] };
      acc[nt] = wmma4(af, bf, acc[nt]);
    }
  }
  const float* bias = a.bias[s];
  const float* res  = a.res[s] ? (a.res[s] + (long)b * a.resB) : nullptr;
  float* dst = a.dst[s] + (long)b * a.dstB[s];
  #pragma unroll
  for (int nt = 0; nt < 4; ++nt)
    #pragma unroll
    for (int r = 0; r < 8; ++r) {
      const int row = m0 + r + (hi ? 8 : 0);
      const int col = nt * 16 + ml;
      float v = tanhf(acc[nt][r] + bias[col]);
      if (res) v += res[(long)row * 64 + col];
      dst[(long)row * 64 + col] = v;
    }
}

// ---------------------------------------------------------------------------
// Generic 64x64 GEMM + activation (Zout stages): out = act(src*W + b)
// ---------------------------------------------------------------------------
__global__ __launch_bounds__(32)
void k_gemm64(const float* __restrict__ src, long sb,
              const float* __restrict__ W, const float* __restrict__ bias,
              float* __restrict__ dst, long db, int act /*0=tanh,1=sigmoid*/)
{
  const int lane = threadIdx.x;
  const int m0 = blockIdx.x * 16;
  const int b  = blockIdx.y;
  const bool hi = lane >= 16;
  const int ml = lane & 15;
  const int koff = hi ? 2 : 0;
  const float* S = src + (long)b * sb;

  v8f acc[4];
  #pragma unroll
  for (int i = 0; i < 4; ++i) acc[i] = vzero8();

  for (int k0 = 0; k0 < 64; k0 += 4) {
    const float* sp = S + (long)(m0 + ml) * 64 + k0 + koff;
    v2f af = { sp[0], sp[1] };
    const float* wp = W + (long)(k0 + koff) * 64 + ml;
    #pragma unroll
    for (int nt = 0; nt < 4; ++nt) {
      v2f bf = { wp[nt * 16], wp[nt * 16 + 64] };
      acc[nt] = wmma4(af, bf, acc[nt]);
    }
  }
  float* D = dst + (long)b * db;
  #pragma unroll
  for (int nt = 0; nt < 4; ++nt)
    #pragma unroll
    for (int r = 0; r < 8; ++r) {
      const int row = m0 + r + (hi ? 8 : 0);
      const int col = nt * 16 + ml;
      float v = acc[nt][r] + bias[col];
      v = act ? sigmoidf_(v) : tanhf(v);
      D[(long)row * 64 + col] = v;
    }
}

// ---------------------------------------------------------------------------
// GRU r/u gates: ru = sigmoid( [x|h]W0 + [Ax|Ah]W1 + [A2x|A2h]W2 + b ),
// h-part (K=192) via WMMA, 2-col x-part folded into epilogue using AX1/AX2.
// Writes u (cols 64..127) and rh = r*h (cols 0..63).
// ---------------------------------------------------------------------------
struct RuArgs {
  const float* h1[2];  long h1B[2];
  const float* Ah[2];
  const float* A2h[2];
  const float* W[2];
  const float* bias[2];
  const float* vals; const float* msk;
  const float* AX1;  const float* AX2;
  float* rh[2];
  float* u[2];
  int t;
};

__global__ __launch_bounds__(32)
void k_ru(RuArgs a)
{
  const int lane = threadIdx.x;
  const int m0 = blockIdx.x * 16;
  const int b  = blockIdx.y;
  const int s  = blockIdx.z;
  const bool hi = lane >= 16;
  const int ml = lane & 15;
  const int koff = hi ? 2 : 0;
  const float* W = a.W[s];
  const float* srcs[3] = { a.h1[s] + (long)b * a.h1B[s],
                           a.Ah[s]  + (long)b * (NN * 64L),
                           a.A2h[s] + (long)b * (NN * 64L) };
  v8f acc[8];
  #pragma unroll
  for (int i = 0; i < 8; ++i) acc[i] = vzero8();

  for (int k0 = 0; k0 < 192; k0 += 4) {
    const int chunk = k0 >> 6, d0 = k0 & 63;
    const float* sp = srcs[chunk] + (long)(m0 + ml) * 64 + d0 + koff;
    v2f af = { sp[0], sp[1] };
    const float* wp = W + (long)chunk * (66 * 128) + (long)(2 + d0 + koff) * 128 + ml;
    #pragma unroll
    for (int nt = 0; nt < 8; ++nt) {
      v2f bf = { wp[nt * 16], wp[nt * 16 + 128] };
      acc[nt] = wmma4(af, bf, acc[nt]);
    }
  }
  const float* bias = a.bias[s];
  const float* h1p = srcs[0];
  float* rhp = a.rh[s] + (long)b * (NN * 64L);
  float* up  = a.u[s]  + (long)b * (NN * 64L);
  #pragma unroll
  for (int r = 0; r < 8; ++r) {
    const int row = m0 + r + (hi ? 8 : 0);
    const long xi = (((long)b * TT + a.t) * NN + row) * 2;
    const float x0 = a.vals[xi] * a.msk[xi];
    const float x1 = a.vals[xi + 1] * a.msk[xi + 1];
    const float* r1 = a.AX1 + (long)b * (NN * 96L) + (long)row * 96 + (long)a.t * 2;
    const float* r2 = a.AX2 + (long)b * (NN * 96L) + (long)row * 96 + (long)a.t * 2;
    const float a10 = r1[0], a11 = r1[1], a20 = r2[0], a21 = r2[1];
    #pragma unroll
    for (int nt = 0; nt < 8; ++nt) {
      const int col = nt * 16 + ml;
      float v = acc[nt][r] + bias[col]
              + x0  * W[col]                 + x1  * W[128 + col]
              + a10 * W[66 * 128 + col]      + a11 * W[66 * 128 + 128 + col]
              + a20 * W[2 * 66 * 128 + col]  + a21 * W[2 * 66 * 128 + 128 + col];
      const float g = sigmoidf_(v);
      if (col < 64) rhp[(long)row * 64 + col] = g * h1p[(long)row * 64 + col];
      else          up[(long)row * 64 + (col - 64)] = g;
    }
  }
}

// ---------------------------------------------------------------------------
// GRU candidate + update + mask blend:
//   c  = tanh([x|rh]W0 + [Ax|Arh]W1 + [A2x|A2rh]W2 + b)
//   h2 = u*h1 + (1-u)*c ;  hn = h1*(1-om) + h2*om  -> state (+ optional out)
// ---------------------------------------------------------------------------
struct CArgs {
  const float* rh[2];
  const float* Arh[2];
  const float* A2rh[2];
  const float* W[2];
  const float* bias[2];
  const float* vals; const float* msk;
  const float* AX1;  const float* AX2;
  const float* h1[2]; long h1B[2];
  const float* u[2];
  const float* om;
  float* st[2];
  float* dup[2]; long dupB;
  int t;
};

__global__ __launch_bounds__(32)
void k_cgate(CArgs a)
{
  const int lane = threadIdx.x;
  const int m0 = blockIdx.x * 16;
  const int b  = blockIdx.y;
  const int s  = blockIdx.z;
  const bool hi = lane >= 16;
  const int ml = lane & 15;
  const int koff = hi ? 2 : 0;
  const float* W = a.W[s];
  const float* srcs[3] = { a.rh[s]   + (long)b * (NN * 64L),
                           a.Arh[s]  + (long)b * (NN * 64L),
                           a.A2rh[s] + (long)b * (NN * 64L) };
  v8f acc[4];
  #pragma unroll
  for (int i = 0; i < 4; ++i) acc[i] = vzero8();

  for (int k0 = 0; k0 < 192; k0 += 4) {
    const int chunk = k0 >> 6, d0 = k0 & 63;
    const float* sp = srcs[chunk] + (long)(m0 + ml) * 64 + d0 + koff;
    v2f af = { sp[0], sp[1] };
    const float* wp = W + (long)chunk * (66 * 64) + (long)(2 + d0 + koff) * 64 + ml;
    #pragma unroll
    for (int nt = 0; nt < 4; ++nt) {
      v2f bf = { wp[nt * 16], wp[nt * 16 + 64] };
      acc[nt] = wmma4(af, bf, acc[nt]);
    }
  }
  const float* bias = a.bias[s];
  const float* h1p = a.h1[s] + (long)b * a.h1B[s];
  const float* up  = a.u[s]  + (long)b * (NN * 64L);
  float* stp = a.st[s] + (long)b * (NN * 64L);
  float* dup = a.dup[s] ? (a.dup[s] + (long)b * a.dupB) : nullptr;
  #pragma unroll
  for (int r = 0; r < 8; ++r) {
    const int row = m0 + r + (hi ? 8 : 0);
    const long xi = (((long)b * TT + a.t) * NN + row) * 2;
    const float x0 = a.vals[xi] * a.msk[xi];
    const float x1 = a.vals[xi + 1] * a.msk[xi + 1];
    const float* r1 = a.AX1 + (long)b * (NN * 96L) + (long)row * 96 + (long)a.t * 2;
    const float* r2 = a.AX2 + (long)b * (NN * 96L) + (long)row * 96 + (long)a.t * 2;
    const float a10 = r1[0], a11 = r1[1], a20 = r2[0], a21 = r2[1];
    const float omv = a.om[((long)b * TT + a.t) * NN + row];
    #pragma unroll
    for (int nt = 0; nt < 4; ++nt) {
      const int col = nt * 16 + ml;
      float v = acc[nt][r] + bias[col]
              + x0  * W[col]                + x1  * W[64 + col]
              + a10 * W[66 * 64 + col]      + a11 * W[66 * 64 + 64 + col]
              + a20 * W[2 * 66 * 64 + col]  + a21 * W[2 * 66 * 64 + 64 + col];
      const float c  = tanhf(v);
      const float uu = up[(long)row * 64 + col];
      const float h1 = h1p[(long)row * 64 + col];
      const float h2 = uu * h1 + (1.0f - uu) * c;
      const float hn = h1 * (1.0f - omv) + h2 * omv;
      stp[(long)row * 64 + col] = hn;
      if (dup) dup[(long)row * 64 + col] = hn;
    }
  }
}

// ---------------------------------------------------------------------------
// Small pointwise kernels
// ---------------------------------------------------------------------------
__global__ void k_om(const float* __restrict__ msk, float* __restrict__ om)
{
  const long i = (long)blockIdx.x * blockDim.x + threadIdx.x;
  if (i >= (long)BB * TT * NN) return;
  const float s = fabsf(msk[i * 2]) + fabsf(msk[i * 2 + 1]);
  om[i] = (s > 1e-4f) ? 1.0f : 0.0f;
}

__global__ void k_init(const float* __restrict__ h0, const float* __restrict__ z0,
                       float* __restrict__ hv, float* __restrict__ zv)
{
  const long i = (long)blockIdx.x * blockDim.x + threadIdx.x;
  if (i >= (long)BB * NN * 64) return;
  const int r = (int)(i & 63);
  hv[i] = h0[r];
  zv[i] = z0[r];
}

__global__ void k_pred(const float* __restrict__ h1, long h1B,
                       const float* __restrict__ gate, long gB,
                       const float* __restrict__ ow, const float* __restrict__ ob,
                       float* __restrict__ xp, int t)
{
  const int idx = blockIdx.x * blockDim.x + threadIdx.x;
  if (idx >= BB * NN) return;
  const int b = idx / NN, n = idx % NN;
  const float* hr = h1   + (long)b * h1B + (long)n * 64;
  const float* gr = gate + (long)b * gB  + (long)n * 64;
  float s0 = 0.f, s1 = 0.f;
  #pragma unroll 8
  for (int d = 0; d < 64; ++d) {
    const float hg = hr[d] * gr[d];
    s0 += hg * ow[d * 2];
    s1 += hg * ow[d * 2 + 1];
  }
  const long o = (((long)b * (TT - 1) + (t - 1)) * NN + n) * 2;
  xp[o]     = s0 + ob[0];
  xp[o + 1] = s1 + ob[1];
}

__global__ void k_traj(const float* __restrict__ tin, float* __restrict__ out)
{
  const int i = threadIdx.x;
  if (i < TT) out[i] = tin[i] + 1.0f;  // DT = 1.0
}

// ---------------------------------------------------------------------------
// Host launcher
// ---------------------------------------------------------------------------
extern "C" void kernel_launch(void* const* d_in, const int* in_sizes, int n_in,
                              void* d_out_, int out_size, void* d_ws, size_t ws_size,
                              hipStream_t stream)
{
  (void)in_sizes; (void)n_in; (void)out_size; (void)ws_size;

  const float* vals = (const float*)d_in[0];
  const float* msk  = (const float*)d_in[1];
  const float* A    = (const float*)d_in[2];
  const float* tin  = (const float*)d_in[3];
  const float* h0   = (const float*)d_in[4];
  const float* z0   = (const float*)d_in[5];
  const float* Wru_v = (const float*)d_in[6];
  const float* bru_v = (const float*)d_in[7];
  const float* Wc_v  = (const float*)d_in[8];
  const float* bc_v  = (const float*)d_in[9];
  const float* Wru_g = (const float*)d_in[10];
  const float* bru_g = (const float*)d_in[11];
  const float* Wc_g  = (const float*)d_in[12];
  const float* bc_g  = (const float*)d_in[13];
  const float* Wov[3] = { (const float*)d_in[14], (const float*)d_in[16], (const float*)d_in[18] };
  const float* bov[3] = { (const float*)d_in[15], (const float*)d_in[17], (const float*)d_in[19] };
  const float* Wog[3] = { (const float*)d_in[20], (const float*)d_in[22], (const float*)d_in[24] };
  const float* bog[3] = { (const float*)d_in[21], (const float*)d_in[23], (const float*)d_in[25] };
  const float* Wz1 = (const float*)d_in[26]; const float* bz1 = (const float*)d_in[27];
  const float* Wz2 = (const float*)d_in[28]; const float* bz2 = (const float*)d_in[29];
  const float* Wz3 = (const float*)d_in[30]; const float* bz3 = (const float*)d_in[31];
  const float* ow  = (const float*)d_in[32]; const float* ob  = (const float*)d_in[33];

  float* out = (float*)d_out_;
  float* ws  = (float*)d_ws;

  // d_out layout (floats): X_pred[B,47,N,2] | hv_pre[B,T,N,R] | hv_post | traj[T]
  const long XPRED = 0;
  const long HPRE  = (long)BB * (TT - 1) * NN * 2;            // 601600
  const long HPOST = HPRE + (long)BB * TT * NN * RR;          // +19660800
  const long TRAJ  = HPOST + (long)BB * TT * NN * RR;

  // workspace layout (floats)
  const long WSB  = (long)NN * 64;      // per-batch state stride  (25600)
  const long AXB  = (long)NN * 96;      // per-batch AX stride     (38400)
  const long OUTB = (long)TT * NN * RR; // per-batch stride in hv_pre/post (1228800)
  long off = 0;
  float* om   = ws + off; off += (long)BB * TT * NN;
  float* AX1  = ws + off; off += (long)BB * AXB;
  float* AX2  = ws + off; off += (long)BB * AXB;
  float* hv   = ws + off; off += (long)BB * WSB;
  float* zv   = ws + off; off += (long)BB * WSB;
  float* zv1  = ws + off; off += (long)BB * WSB;
  float* Y1v  = ws + off; off += (long)BB * WSB;
  float* Y1g  = ws + off; off += (long)BB * WSB;
  float* Y2v  = ws + off; off += (long)BB * WSB;
  float* Y2g  = ws + off; off += (long)BB * WSB;
  float* yAv  = ws + off; off += (long)BB * WSB;
  float* yAg  = ws + off; off += (long)BB * WSB;
  float* yBv  = ws + off; off += (long)BB * WSB;
  float* yBg  = ws + off; off += (long)BB * WSB;
  float* Ahv  = ws + off; off += (long)BB * WSB;
  float* Ahg  = ws + off; off += (long)BB * WSB;
  float* A2hv = ws + off; off += (long)BB * WSB;
  float* A2hg = ws + off; off += (long)BB * WSB;
  float* rhv  = ws + off; off += (long)BB * WSB;
  float* rhg  = ws + off; off += (long)BB * WSB;
  float* Arhv = ws + off; off += (long)BB * WSB;
  float* Arhg = ws + off; off += (long)BB * WSB;
  float* A2rhv= ws + off; off += (long)BB * WSB;
  float* A2rhg= ws + off; off += (long)BB * WSB;
  float* uv   = ws + off; off += (long)BB * WSB;
  float* ug   = ws + off; off += (long)BB * WSB;
  float* g1   = ws + off; off += (long)BB * WSB;
  float* g2   = ws + off; off += (long)BB * WSB;

  const dim3 blk(32, 1, 1);
  const dim3 g2s(NN / 16, BB, 2);   // both states
  const dim3 g1s(NN / 16, BB, 1);

  // ---- Phase 0: masks, init states, precompute A*x / A^2*x for all t ----
  k_om  <<<(BB * TT * NN + 255) / 256, 256, 0, stream>>>(msk, om);
  k_init<<<(BB * NN * 64 + 255) / 256, 256, 0, stream>>>(h0, z0, hv, zv);
  k_ax_input<<<g1s, blk, 0, stream>>>(A, vals, msk, AX1);
  k_diffuse<96><<<g1s, blk, 0, stream>>>(A, AX1, AXB, AX1, AXB, AX2, AXB, AX2, AXB);
  k_traj<<<1, 64, 0, stream>>>(tin, out + TRAJ);

  // ---- Phase 1: sequential scan over T ----
  for (int t = 0; t < TT; ++t) {
    float* hpre  = out + HPRE  + (long)t * WSB;   // per-b stride OUTB
    float* hpost = out + HPOST + (long)t * WSB;

    // ODE layer 0
    k_diffuse<64><<<g2s, blk, 0, stream>>>(A, hv, WSB, zv, WSB, Y1v, WSB, Y1g, WSB);
    k_diffuse<64><<<g2s, blk, 0, stream>>>(A, Y1v, WSB, Y1g, WSB, Y2v, WSB, Y2g, WSB);
    {
      L3Args a;
      a.src[0][0]=hv;  a.src[0][1]=Y1v; a.src[0][2]=Y2v;
      a.src[1][0]=zv;  a.src[1][1]=Y1g; a.src[1][2]=Y2g;
      a.W[0]=Wov[0]; a.W[1]=Wog[0]; a.bias[0]=bov[0]; a.bias[1]=bog[0];
      a.res[0]=nullptr; a.res[1]=nullptr; a.resB=WSB;
      a.dst[0]=yAv; a.dst[1]=yAg; a.dstB[0]=WSB; a.dstB[1]=WSB; a.srcB=WSB;
      k_layer3<<<g2s, blk, 0, stream>>>(a);
    }
    // ODE layer 1
    k_diffuse<64><<<g2s, blk, 0, stream>>>(A, yAv, WSB, yAg, WSB, Y1v, WSB, Y1g, WSB);
    k_diffuse<64><<<g2s, blk, 0, stream>>>(A, Y1v, WSB, Y1g, WSB, Y2v, WSB, Y2g, WSB);
    {
      L3Args a;
      a.src[0][0]=yAv; a.src[0][1]=Y1v; a.src[0][2]=Y2v;
      a.src[1][0]=yAg; a.src[1][1]=Y1g; a.src[1][2]=Y2g;
      a.W[0]=Wov[1]; a.W[1]=Wog[1]; a.bias[0]=bov[1]; a.bias[1]=bog[1];
      a.res[0]=nullptr; a.res[1]=nullptr; a.resB=WSB;
      a.dst[0]=yBv; a.dst[1]=yBg; a.dstB[0]=WSB; a.dstB[1]=WSB; a.srcB=WSB;
      k_layer3<<<g2s, blk, 0, stream>>>(a);
    }
    // ODE layer 2 (+ residual) -> hv1 into d_out hv_pre, zv1 into ws
    k_diffuse<64><<<g2s, blk, 0, stream>>>(A, yBv, WSB, yBg, WSB, Y1v, WSB, Y1g, WSB);
    k_diffuse<64><<<g2s, blk, 0, stream>>>(A, Y1v, WSB, Y1g, WSB, Y2v, WSB, Y2g, WSB);
    {
      L3Args a;
      a.src[0][0]=yBv; a.src[0][1]=Y1v; a.src[0][2]=Y2v;
      a.src[1][0]=yBg; a.src[1][1]=Y1g; a.src[1][2]=Y2g;
      a.W[0]=Wov[2]; a.W[1]=Wog[2]; a.bias[0]=bov[2]; a.bias[1]=bog[2];
      a.res[0]=hv; a.res[1]=zv; a.resB=WSB;
      a.dst[0]=hpre; a.dst[1]=zv1; a.dstB[0]=OUTB; a.dstB[1]=WSB; a.srcB=WSB;
      k_layer3<<<g2s, blk, 0, stream>>>(a);
    }

    // Zout gate chain on zv1 (this step's zv_pre) + prediction
    k_gemm64<<<g1s, blk, 0, stream>>>(zv1, WSB, Wz1, bz1, g1, WSB, 0);
    k_gemm64<<<g1s, blk, 0, stream>>>(g1,  WSB, Wz2, bz2, g2, WSB, 0);
    k_gemm64<<<g1s, blk, 0, stream>>>(g2,  WSB, Wz3, bz3, g1, WSB, 1);
    if (t > 0)
      k_pred<<<(BB * NN + 255) / 256, 256, 0, stream>>>(hpre, OUTB, g1, WSB,
                                                        ow, ob, out + XPRED, t);

    // GRU: diffusion of h1, r/u gates
    k_diffuse<64><<<g2s, blk, 0, stream>>>(A, hpre, OUTB, zv1, WSB, Ahv, WSB, Ahg, WSB);
    k_diffuse<64><<<g2s, blk, 0, stream>>>(A, Ahv, WSB, Ahg, WSB, A2hv, WSB, A2hg, WSB);
    {
      RuArgs a;
      a.h1[0]=hpre; a.h1[1]=zv1; a.h1B[0]=OUTB; a.h1B[1]=WSB;
      a.Ah[0]=Ahv;  a.Ah[1]=Ahg; a.A2h[0]=A2hv; a.A2h[1]=A2hg;
      a.W[0]=Wru_v; a.W[1]=Wru_g; a.bias[0]=bru_v; a.bias[1]=bru_g;
      a.vals=vals; a.msk=msk; a.AX1=AX1; a.AX2=AX2;
      a.rh[0]=rhv; a.rh[1]=rhg; a.u[0]=uv; a.u[1]=ug; a.t=t;
      k_ru<<<g2s, blk, 0, stream>>>(a);
    }
    // GRU: diffusion of r*h, candidate + update + blend
    k_diffuse<64><<<g2s, blk, 0, stream>>>(A, rhv, WSB, rhg, WSB, Arhv, WSB, Arhg, WSB);
    k_diffuse<64><<<g2s, blk, 0, stream>>>(A, Arhv, WSB, Arhg, WSB, A2rhv, WSB, A2rhg, WSB);
    {
      CArgs a;
      a.rh[0]=rhv; a.rh[1]=rhg; a.Arh[0]=Arhv; a.Arh[1]=Arhg;
      a.A2rh[0]=A2rhv; a.A2rh[1]=A2rhg;
      a.W[0]=Wc_v; a.W[1]=Wc_g; a.bias[0]=bc_v; a.bias[1]=bc_g;
      a.vals=vals; a.msk=msk; a.AX1=AX1; a.AX2=AX2;
      a.h1[0]=hpre; a.h1[1]=zv1; a.h1B[0]=OUTB; a.h1B[1]=WSB;
      a.u[0]=uv; a.u[1]=ug; a.om=om;
      a.st[0]=hv; a.st[1]=zv;
      a.dup[0]=hpost; a.dup[1]=nullptr; a.dupB=OUTB;
      a.t=t;
      k_cgate<<<g2s, blk, 0, stream>>>(a);
    }
  }
}